// XSReLU_cw_perc_param_47528108097997
// MI455X (gfx1250) — compile-verified
//
#include <hip/hip_runtime.h>

// XSReLU (channel-wise percentile threshold) for MI455X / gfx1250.
//
// Per row (B*C = 8192 rows of HW = 4096 fp32):
//   - TDM tensor_load_to_lds stages the 16KB row into LDS (TENSORcnt),
//     fallback: GLOBAL_LOAD_ASYNC_TO_LDS_B128 (ASYNCcnt)
//   - each thread caches its 16 elements in VGPRs as order-preserving uints
//   - fused dual radix-select (8-bit MSD digits) with early termination:
//     typical path = pass0 + pass1 + one fetch scan, exact vs jnp.sort+take
//   - digit search: wave32 __shfl_up scan + 8 wave totals (1 barrier)
//   - thr = x_low + (x_high - x_low) * sigmoid(plogit[c]); relu written with
//     b128 stores straight from the register copy
//
// Memory-bound target: 128MB read + 128MB write => ~11us @ 23.3 TB/s.

#define N_HW   4096
#define NTHR   256
#define EPT    16          // elements per thread
#define CHUNKS 4           // float4 chunks per thread

typedef unsigned int v4u __attribute__((ext_vector_type(4)));
typedef int          v8i __attribute__((ext_vector_type(8)));
typedef int          v4i __attribute__((ext_vector_type(4)));

__device__ __forceinline__ unsigned f2ord(float f) {
  unsigned u = __float_as_uint(f);
  return (u & 0x80000000u) ? ~u : (u | 0x80000000u);
}
__device__ __forceinline__ float ord2f(unsigned o) {
  unsigned u = (o & 0x80000000u) ? (o ^ 0x80000000u) : ~o;
  return __uint_as_float(u);
}

struct Sel {                 // uniform across the workgroup
  int k;                     // remaining rank
  unsigned prefix, mask;     // resolved high bits
  unsigned val;              // final ordered-uint value
  bool done, fetch;
};

// 256-bin digit search: wave32 shfl inclusive scan + wave-total combine.
// Must be called by all 256 threads (callers guard with *uniform* flags).
__device__ __forceinline__ void find_digit(const unsigned* hist_t, unsigned* bc,
                                           unsigned k, unsigned* wsum, int tid) {
  unsigned h = hist_t[tid];
  unsigned v = h;
#pragma unroll
  for (int o = 1; o < 32; o <<= 1) {
    unsigned n = __shfl_up(v, o, 32);
    if ((tid & 31) >= o) v += n;
  }
  if ((tid & 31) == 31) wsum[tid >> 5] = v;
  __syncthreads();
  unsigned base = 0;
  for (int w = 0; w < (tid >> 5); ++w) base += wsum[w];
  unsigned hi = base + v;
  unsigned lo = hi - h;
  if (k >= lo && k < hi) { bc[0] = (unsigned)tid; bc[1] = lo; bc[2] = h; }
  __syncthreads();
}

__device__ __forceinline__ void apply_digit(Sel& t, const unsigned* bc,
                                            int shift, int pass) {
  unsigned d = bc[0], lo = bc[1], cnt = bc[2];
  t.prefix |= d << shift;
  t.mask   |= 0xFFu << shift;
  t.k      -= (int)lo;
  if (pass == 3)      { t.val  = t.prefix; t.done = true; }  // all 32 bits known
  else if (cnt == 1u) { t.fetch = true;    t.done = true; }  // unique candidate
}

__global__ void __launch_bounds__(NTHR)
xsrelu_cw_perc_kernel(const float* __restrict__ x,
                      const float* __restrict__ plogit,
                      float* __restrict__ out, int C) {
  __shared__ __align__(16) float smem[N_HW];
  __shared__ unsigned hist[2][256];
  __shared__ unsigned bc[2][3];      // per target: digit, count-below, bin-count
  __shared__ unsigned wsum[8];

  const int tid = threadIdx.x;
  const int row = blockIdx.x;              // row = b*C + c
  const int c   = row % C;
  const float* __restrict__ g = x + (size_t)row * N_HW;
  // flat LDS pointer: low 32 bits are the LDS byte offset (aperture rule)
  const unsigned lds_base = (unsigned)(unsigned long long)(const void*)smem;

  // ---- stage row -> LDS -------------------------------------------------
#if __has_builtin(__builtin_amdgcn_tensor_load_to_lds)
  if (tid < 32) {                          // one wave issues the DMA
    unsigned long long ga = (unsigned long long)g;
    v4u g0;                                // D# group 0
    g0.x = 1u;                             // count=1, user descriptor
    g0.y = lds_base;                       // lds_addr
    g0.z = (unsigned)(ga & 0xFFFFFFFFull);           // global_addr[31:0]
    g0.w = (unsigned)((ga >> 32) & 0x01FFFFFFull)    // global_addr[56:32]
         | (2u << 30);                               // type=2 ("image")
    v8i g1;                                // D# group 1
    g1[0] = (int)(2u << 16);               // data_size=2 (4 bytes)
    g1[1] = (int)(0x1000u << 16);          // tensor_dim0 = 4096
    g1[2] = (int)(1u << 16);               // tensor_dim1 = 1
    g1[3] = (int)(0x1000u << 16);          // tile_dim0   = 4096
    g1[4] = 1;                             // tile_dim1=1, tile_dim2=0
    g1[5] = 0x1000;                        // tensor_dim0_stride = 4096
    g1[6] = 0;                             // stride hi / dim1_stride = 0
    g1[7] = 0;
    v4i gz4 = {0, 0, 0, 0};                // groups 2/3 unused (<=2D tensor)
    v8i gz8 = {0, 0, 0, 0, 0, 0, 0, 0};    // extra group (6-arg toolchain form)
    __builtin_amdgcn_tensor_load_to_lds(g0, g1, gz4, gz4, gz8, 0);
    __builtin_amdgcn_s_wait_tensorcnt(0);  // drain TENSORcnt
  }
  __syncthreads();                         // LDS visible workgroup-wide
#else
#pragma unroll
  for (int kk = 0; kk < CHUNKS; ++kk) {    // 4 rounds of 4KB async-to-LDS
    int q = tid + NTHR * kk;
    const float* ga = g + q * 4;
    unsigned la = lds_base + (unsigned)q * 16u;
    asm volatile("global_load_async_to_lds_b128 %0, %1, off"
                 :: "v"(la), "v"(ga) : "memory");
  }
  asm volatile("s_wait_asynccnt 0" ::: "memory");
  __syncthreads();
#endif

  // ---- cache my 16 elements in registers (floats + ordered uints) -------
  const float4* s4 = (const float4*)smem;
  float4 vals[CHUNKS];
  unsigned uv[EPT];
#pragma unroll
  for (int kk = 0; kk < CHUNKS; ++kk) {
    int q = tid + NTHR * kk;               // b128 chunk id (coalesced)
    vals[kk] = s4[q];                      // ds_load_b128
    uv[4 * kk + 0] = f2ord(vals[kk].x);
    uv[4 * kk + 1] = f2ord(vals[kk].y);
    uv[4 * kk + 2] = f2ord(vals[kk].z);
    uv[4 * kk + 3] = f2ord(vals[kk].w);
  }

  // ---- scalar rank indices from plogit[0] (trunc-toward-zero + clamp) ---
  float p0 = 1.0f / (1.0f + expf(-plogit[0]));
  int i_low  = min(max((int)((float)N_HW * (p0 - 0.02f)), 0), N_HW - 1);
  int i_high = min(max((int)((float)N_HW * (p0 + 0.02f)), 0), N_HW - 1);

  Sel t0 = { i_low,  0u, 0u, 0u, false, false };
  Sel t1 = { i_high, 0u, 0u, 0u, false, false };

  // ---- pass 0: one histogram serves both ranks --------------------------
  hist[0][tid] = 0u;
  __syncthreads();
#pragma unroll
  for (int e = 0; e < EPT; ++e)
    atomicAdd(&hist[0][uv[e] >> 24], 1u);  // ds_add_u32
  __syncthreads();
  find_digit(hist[0], bc[0], (unsigned)t0.k, wsum, tid);
  find_digit(hist[0], bc[1], (unsigned)t1.k, wsum, tid);
  apply_digit(t0, bc[0], 24, 0);
  apply_digit(t1, bc[1], 24, 0);

  // ---- passes 1..3: fused dual histogram, register-resident scan --------
  for (int pass = 1; pass < 4 && (!t0.done || !t1.done); ++pass) {
    const int shift = 24 - 8 * pass;
    if (!t0.done) hist[0][tid] = 0u;
    if (!t1.done) hist[1][tid] = 0u;
    __syncthreads();
#pragma unroll
    for (int e = 0; e < EPT; ++e) {
      unsigned u = uv[e];
      if (!t0.done && (u & t0.mask) == t0.prefix)
        atomicAdd(&hist[0][(u >> shift) & 0xFFu], 1u);
      if (!t1.done && (u & t1.mask) == t1.prefix)
        atomicAdd(&hist[1][(u >> shift) & 0xFFu], 1u);
    }
    __syncthreads();
    if (!t0.done) { find_digit(hist[0], bc[0], (unsigned)t0.k, wsum, tid);
                    apply_digit(t0, bc[0], shift, pass); }
    if (!t1.done) { find_digit(hist[1], bc[1], (unsigned)t1.k, wsum, tid);
                    apply_digit(t1, bc[1], shift, pass); }
  }

  // ---- fetch scan: resolve unique candidates to full 32-bit values ------
  if (t0.fetch || t1.fetch) {
    __syncthreads();                       // protect bc from in-flight reads
#pragma unroll
    for (int e = 0; e < EPT; ++e) {
      unsigned u = uv[e];
      if (t0.fetch && (u & t0.mask) == t0.prefix) bc[0][0] = u;  // 1 writer
      if (t1.fetch && (u & t1.mask) == t1.prefix) bc[1][0] = u;  // 1 writer
    }
    __syncthreads();
    if (t0.fetch) t0.val = bc[0][0];
    if (t1.fetch) t1.val = bc[1][0];
  }

  // ---- threshold + relu, streamed from registers with b128 stores -------
  float x_low  = ord2f(t0.val);
  float x_high = ord2f(t1.val);
  float pc  = 1.0f / (1.0f + expf(-plogit[c]));
  float thr = x_low + (x_high - x_low) * pc;

  float4* o4 = (float4*)(out + (size_t)row * N_HW);
#pragma unroll
  for (int kk = 0; kk < CHUNKS; ++kk) {
    int q = tid + NTHR * kk;
    float4 v = vals[kk];
    v.x = fmaxf(v.x - thr, 0.0f);
    v.y = fmaxf(v.y - thr, 0.0f);
    v.z = fmaxf(v.z - thr, 0.0f);
    v.w = fmaxf(v.w - thr, 0.0f);
    o4[q] = v;
  }
}

extern "C" void kernel_launch(void* const* d_in, const int* in_sizes, int n_in,
                              void* d_out, int out_size, void* d_ws, size_t ws_size,
                              hipStream_t stream) {
  const float* x      = (const float*)d_in[0];   // [B,C,H,W] fp32
  const float* plogit = (const float*)d_in[1];   // [C] fp32
  float* out          = (float*)d_out;           // [B,C,H,W] fp32

  const int C    = in_sizes[1];                  // 256
  const int rows = in_sizes[0] / N_HW;           // B*C = 8192

  xsrelu_cw_perc_kernel<<<dim3(rows), dim3(NTHR), 0, stream>>>(x, plogit, out, C);
}